// Layer_Y_81965155877406
// MI455X (gfx1250) — compile-verified
//
#include <hip/hip_runtime.h>
#include <hip/hip_bf16.h>
#include <math.h>

// Memory-bound elementwise kernel (l=3 real spherical harmonics / r^3).
// 800 MB of HBM traffic, ~0.9 GFLOP -> HBM roofline ~34us @ 23.3 TB/s.
// No matmul structure -> WMMA inapplicable; optimize the data path instead:
// async global<->LDS B128 staging (gfx1250 ASYNCcnt path), NT temporal hints
// (read-once/write-once 800MB stream vs 192MB L2), with a portable
// B128 + nontemporal fallback.

typedef __attribute__((ext_vector_type(4))) float v4f;
typedef __attribute__((ext_vector_type(4))) int   v4i;

// Builtin signature (verified by round-1 diagnostics + round-2 codegen):
// (addrspace(1) int4*, addrspace(3) int4*, imm offset, imm cpol)
// -> lowers to global_load_async_to_lds_b128 / global_store_async_from_lds_b128.
#define ASYNC_GPTR(p) ((__attribute__((address_space(1))) v4i*)(p))
#define ASYNC_LPTR(p) ((__attribute__((address_space(3))) v4i*)(p))

// gfx12+ CPol: TH in bits [2:0]; TH=1 is NT for both loads and stores.
#define CPOL_NT 1

#define HAVE_GFX1250_ASYNC                                          \
  (__has_builtin(__builtin_amdgcn_global_load_async_to_lds_b128) && \
   __has_builtin(__builtin_amdgcn_global_store_async_from_lds_b128))

// 256 threads, 256 atoms per block.
// in : 256*3 = 768 floats  = 192 float4
// out: 256*7 = 1792 floats = 448 float4
__global__ __launch_bounds__(256) void
sh_l3_kernel(const float* __restrict__ in, float* __restrict__ out,
             long long n_in4, long long n_out4) {
  __shared__ __attribute__((aligned(16))) float lin[768];
  __shared__ __attribute__((aligned(16))) float lout[1792];

  const int tid = threadIdx.x;
  const long long b = blockIdx.x;
  const v4f* gin4  = reinterpret_cast<const v4f*>(in) + b * 192;
  v4f*       gout4 = reinterpret_cast<v4f*>(out)      + b * 448;
  const long long gi4  = b * 192 + tid;
  const long long go4a = b * 448 + tid;
  const long long go4b = go4a + 256;

  // ---- Stage input block into LDS with coalesced async B128 traffic ----
#if HAVE_GFX1250_ASYNC
  if (tid < 192 && gi4 < n_in4) {
    __builtin_amdgcn_global_load_async_to_lds_b128(
        ASYNC_GPTR(gin4 + tid),
        ASYNC_LPTR(&lin[tid * 4]),
        /*offset=*/0, /*cpol=*/CPOL_NT);
  }
#if __has_builtin(__builtin_amdgcn_s_wait_asynccnt)
  __builtin_amdgcn_s_wait_asynccnt(0);
#else
  asm volatile("s_wait_asynccnt 0" ::: "memory");
#endif
  __syncthreads();
#else
  if (tid < 192 && gi4 < n_in4) {
    v4f v = __builtin_nontemporal_load(gin4 + tid);  // streamed once
    *reinterpret_cast<v4f*>(&lin[tid * 4]) = v;
  }
  __syncthreads();
#endif

  // ---- Compute: one atom per thread ----
  // lin read stride 3 words, lout write stride 7 words: both coprime with the
  // 64-bank LDS -> conflict-free.
  {
    const float x = lin[tid * 3 + 0];
    const float y = lin[tid * 3 + 1];
    const float z = lin[tid * 3 + 2];
    const float x2 = x * x, y2 = y * y, z2 = z * z;
    const float x2my2 = x2 - y2;
    const float r2 = x2 + y2 + z2;
    const float inv_r3 = 1.0f / (sqrtf(r2) * r2);
    float* o = &lout[tid * 7];
    o[0] = (3.0f * x2 - y2) * y * inv_r3;
    o[1] = x * y * z * inv_r3;
    o[2] = (4.0f * z2 - x2my2) * y * inv_r3;
    o[3] = (2.0f * z2 - 3.0f * x2my2) * z * inv_r3;
    o[4] = (4.0f * z2 - x2my2) * x * inv_r3;
    o[5] = z * x2my2 * inv_r3;
    o[6] = x * (x2 - 3.0f * y2) * inv_r3;
  }
  __syncthreads();

  // ---- Stream output block to global with full-cacheline B128 stores ----
#if HAVE_GFX1250_ASYNC
  if (go4a < n_out4) {
    __builtin_amdgcn_global_store_async_from_lds_b128(
        ASYNC_GPTR(gout4 + tid),
        ASYNC_LPTR(&lout[tid * 4]),
        /*offset=*/0, /*cpol=*/CPOL_NT);
  }
  if (tid < 192 && go4b < n_out4) {
    __builtin_amdgcn_global_store_async_from_lds_b128(
        ASYNC_GPTR(gout4 + 256 + tid),
        ASYNC_LPTR(&lout[(256 + tid) * 4]),
        /*offset=*/0, /*cpol=*/CPOL_NT);
  }
  // S_ENDPGM performs an implicit wait-idle, draining ASYNCcnt before the
  // wave retires (and before LDS deallocation), so stores complete safely.
#else
  if (go4a < n_out4) {
    v4f a = *reinterpret_cast<v4f*>(&lout[tid * 4]);
    __builtin_nontemporal_store(a, gout4 + tid);  // written once, never reread
  }
  if (tid < 192 && go4b < n_out4) {
    v4f c = *reinterpret_cast<v4f*>(&lout[(256 + tid) * 4]);
    __builtin_nontemporal_store(c, gout4 + 256 + tid);
  }
#endif
}

extern "C" void kernel_launch(void* const* d_in, const int* in_sizes, int n_in,
                              void* d_out, int out_size, void* d_ws, size_t ws_size,
                              hipStream_t stream) {
  (void)n_in; (void)d_ws; (void)ws_size;
  const float* in = (const float*)d_in[0];
  float* out = (float*)d_out;

  const long long n_floats = (long long)in_sizes[0];   // 60,000,000
  const long long n_atoms  = n_floats / 3;             // 20,000,000 = 256 * 78125
  const long long n_in4    = n_floats / 4;             // 15,000,000
  const long long n_out4   = (long long)out_size / 4;  // 35,000,000

  const int blocks = (int)((n_atoms + 255) / 256);     // 78,125 (exact fit)
  sh_l3_kernel<<<blocks, 256, 0, stream>>>(in, out, n_in4, n_out4);
}